// QuantumEncoder_8693013807449
// MI455X (gfx1250) — compile-verified
//
#include <hip/hip_runtime.h>

typedef float v2f __attribute__((ext_vector_type(2)));
typedef float v8f __attribute__((ext_vector_type(8)));

#define NQ 4
#define DIM 16           // 2^NQ amplitudes
#define WAVES_PER_BLOCK 8

// ---------------------------------------------------------------------------
// Kernel 1: build V = U * diag((-i)^popcount) from the shared weights.
// U = (2 layers of: RX(weights[l][w]) on each wire, then ring CNOTs).
// 16 threads, thread t owns column t of U (starts as e_t), gates applied from
// the left entirely in registers (all loops unrolled -> constant indexing).
// Wire w <-> bit (3-w)  (wire 0 is the most-significant axis in the reference).
// Output: Vr[j*16+k], Vi[j*16+k] row-major (j = output basis, k = input basis).
// ---------------------------------------------------------------------------
__global__ void build_V_kernel(const float* __restrict__ weights,
                               float* __restrict__ Vr,
                               float* __restrict__ Vi) {
    int t = threadIdx.x;
    if (t >= DIM) return;

    float ur[DIM], ui[DIM];
#pragma unroll
    for (int j = 0; j < DIM; ++j) { ur[j] = (j == t) ? 1.0f : 0.0f; ui[j] = 0.0f; }

#pragma unroll
    for (int l = 0; l < 2; ++l) {
        // RX(weights[l][w]) on each wire: u' = G u, G = [[c, -is], [-is, c]]
#pragma unroll
        for (int w = 0; w < NQ; ++w) {
            float th = 0.5f * weights[l * NQ + w];
            float c = __cosf(th);
            float s = __sinf(th);
            const int mask = 1 << (3 - w);
#pragma unroll
            for (int a = 0; a < DIM; ++a) {
                if ((a & mask) == 0) {
                    const int b = a | mask;
                    float ar = ur[a], ai = ui[a], br = ur[b], bi = ui[b];
                    // new_a = c*a + (-i s)*b ; new_b = (-i s)*a + c*b
                    ur[a] = c * ar + s * bi;  ui[a] = c * ai - s * br;
                    ur[b] = c * br + s * ai;  ui[b] = c * bi - s * ar;
                }
            }
        }
        // Ring CNOTs: (0,1),(1,2),(2,3),(3,0). ctrl bit set -> flip target bit.
#pragma unroll
        for (int w = 0; w < NQ; ++w) {
            const int cm = 1 << (3 - w);
            const int tm = 1 << (3 - ((w + 1) & 3));
            float nr[DIM], ni[DIM];
#pragma unroll
            for (int a = 0; a < DIM; ++a) {
                const int src = (a & cm) ? (a ^ tm) : a;
                nr[a] = ur[src]; ni[a] = ui[src];
            }
#pragma unroll
            for (int a = 0; a < DIM; ++a) { ur[a] = nr[a]; ui[a] = ni[a]; }
        }
    }

    // Fold column phase (-i)^p, p = popcount(t):
    //   (-i)^p = sigma * (p odd ? -i : 1),  sigma = (p mod 4 < 2) ? +1 : -1
    const int p = __popc(t);
    const float sg = ((p & 2) == 0) ? 1.0f : -1.0f;
#pragma unroll
    for (int j = 0; j < DIM; ++j) {
        float vr, vi;
        if (p & 1) { vr = sg * ui[j]; vi = -sg * ur[j]; }  // (a+bi)(-i) = b - ai
        else       { vr = sg * ur[j]; vi =  sg * ui[j]; }
        Vr[j * DIM + t] = vr;
        Vi[j * DIM + t] = vi;
    }
}

// ---------------------------------------------------------------------------
// Kernel 2: per wave, a 16-sample tile.
//   A[m,k] = rho[m,k]  (real reduced state; product of cos/sin half-angles)
//   B_r = Vr^T, B_i = Vi^T  ->  phi_r = A*B_r, phi_i = A*B_i   (K = 16)
//   4 + 4  v_wmma_f32_16x16x4_f32  accumulate phi_r / phi_i.
//   Then p = phi_r^2 + phi_i^2 -> LDS -> signed Z reduction -> out[m][w].
//
// f32 WMMA layouts (ISA 7.12.2):
//   A 16x4 : lane L, reg v ->  M = L%16, K = 2*(L/16) + v
//   B 4x16 : lane L, reg v ->  N = L%16, K = 2*(L/16) + v   (mirror of A)
//   C/D    : lane L, reg r ->  N = L%16, M = r + 8*(L/16)
// ---------------------------------------------------------------------------
__global__ __launch_bounds__(32 * WAVES_PER_BLOCK)
void qsim_wmma_kernel(const float* __restrict__ x,
                      const float* __restrict__ Vr,
                      const float* __restrict__ Vi,
                      float* __restrict__ out,
                      int nsamp) {
    __shared__ float pl[WAVES_PER_BLOCK][DIM * DIM];

    const int lane = threadIdx.x & 31;
    const int wv   = threadIdx.x >> 5;
    const int half = lane >> 4;        // 0 or 1
    const int lm   = lane & 15;        // row (M) for A, col (N) for B/D

    const long tile = (long)blockIdx.x * WAVES_PER_BLOCK + wv;
    const long m0   = tile * DIM;

    // ---- B fragments (sample independent): chunk c covers K = 4c..4c+3 ----
    v2f br[4], bi[4];
#pragma unroll
    for (int c = 0; c < 4; ++c) {
        const int k0 = c * 4 + half * 2;
        br[c][0] = Vr[lm * DIM + k0];      // B_r[k,n] = Vr[n][k]
        br[c][1] = Vr[lm * DIM + k0 + 1];
        bi[c][0] = Vi[lm * DIM + k0];
        bi[c][1] = Vi[lm * DIM + k0 + 1];
    }

    // ---- A fragments: rho for sample m = m0 + lm (clamped load, no branch) ----
    long m = m0 + lm;
    long mc = (m < nsamp) ? m : (long)(nsamp - 1);
    float cw[NQ], sw[NQ];
#pragma unroll
    for (int w = 0; w < NQ; ++w) {
        float th = 0.5f * x[mc * NQ + w];
        cw[w] = __cosf(th);
        sw[w] = __sinf(th);
    }
    v2f a[4];
#pragma unroll
    for (int c = 0; c < 4; ++c) {
#pragma unroll
        for (int v = 0; v < 2; ++v) {
            const int k = c * 4 + half * 2 + v;   // K index 0..15
            float r = 1.0f;
#pragma unroll
            for (int w = 0; w < NQ; ++w)
                r *= ((k >> (3 - w)) & 1) ? sw[w] : cw[w];
            a[c][v] = r;
        }
    }

    // ---- GEMM: phi_r, phi_i (uniform flow, EXEC all ones) ----
    v8f dr = {};
    v8f di = {};
#pragma unroll
    for (int c = 0; c < 4; ++c) {
        dr = __builtin_amdgcn_wmma_f32_16x16x4_f32(false, a[c], false, br[c],
                                                   (short)0, dr, false, false);
        di = __builtin_amdgcn_wmma_f32_16x16x4_f32(false, a[c], false, bi[c],
                                                   (short)0, di, false, false);
    }

    // ---- probabilities into LDS tile (sample-major) ----
#pragma unroll
    for (int r = 0; r < 8; ++r) {
        const int mm = r + 8 * half;              // sample row 0..15
        pl[wv][mm * DIM + lm] = dr[r] * dr[r] + di[r] * di[r];
    }
    __syncthreads();

    // ---- signed Z reduction: 64 outputs per tile (16 samples x 4 wires) ----
#pragma unroll
    for (int t = 0; t < 2; ++t) {
        const int idx = lane + t * 32;
        const int mm  = idx >> 2;                 // sample 0..15
        const int w   = idx & 3;                  // wire 0..3
        const int sh  = 3 - w;
        float acc = 0.0f;
#pragma unroll
        for (int n = 0; n < DIM; ++n) {
            const float v = pl[wv][mm * DIM + n];
            acc += ((n >> sh) & 1) ? -v : v;
        }
        const long ms = m0 + mm;
        if (ms < nsamp) out[ms * NQ + w] = acc;
    }
}

// ---------------------------------------------------------------------------
extern "C" void kernel_launch(void* const* d_in, const int* in_sizes, int n_in,
                              void* d_out, int out_size, void* d_ws, size_t ws_size,
                              hipStream_t stream) {
    const float* x       = (const float*)d_in[0];   // [B, 4] float32
    const float* weights = (const float*)d_in[1];   // [2, 4] float32
    float* out = (float*)d_out;                     // [B, 4] float32

    float* Vr = (float*)d_ws;                       // 256 floats
    float* Vi = Vr + DIM * DIM;                     // 256 floats

    const int nsamp = in_sizes[0] / NQ;

    build_V_kernel<<<1, 32, 0, stream>>>(weights, Vr, Vi);

    const int tiles  = (nsamp + DIM - 1) / DIM;
    const int blocks = (tiles + WAVES_PER_BLOCK - 1) / WAVES_PER_BLOCK;
    qsim_wmma_kernel<<<blocks, 32 * WAVES_PER_BLOCK, 0, stream>>>(x, Vr, Vi, out, nsamp);
}